// CircularConvolutionHelper_4947802325454
// MI455X (gfx1250) — compile-verified
//
#include <hip/hip_runtime.h>

// CDNA5 / gfx1250: wave32. Pipeline: HBM/L2 -> TDM (tensor_load_to_lds) -> LDS
// -> V_WMMA_F32_16X16X4_F32 -> ds_add_f32 diagonal reduce -> global_atomic_add.
typedef __attribute__((ext_vector_type(2))) float v2f;
typedef __attribute__((ext_vector_type(8))) float v8f;
typedef unsigned int v4u __attribute__((ext_vector_type(4)));
typedef int          v4i __attribute__((ext_vector_type(4)));
typedef int          v8i __attribute__((ext_vector_type(8)));

#define DIM   1024
#define BATCH 128

// Harness poisons d_out with 0xAA; we accumulate with atomics, so zero first.
__global__ void cc_zero_out(float* __restrict__ out) {
    const int t = threadIdx.x;
#pragma unroll
    for (int i = 0; i < 4; ++i) out[t * 4 + i] = 0.0f;
}

// Build a 2D Tensor-DMA descriptor (ISA ch.8 D# layout) and issue
// TENSOR_LOAD_TO_LDS. data_size = 4 bytes. Groups 2/3 unused (2D tensor).
// This toolchain's builtin takes 6 args (g0, g1, g2, g3, g4, cpol).
__device__ __forceinline__ void tdm_load_2d(unsigned int lds_off, const void* gptr,
                                            unsigned int tile_d0, unsigned int tile_d1,
                                            unsigned int tensor_d0, unsigned int tensor_d1,
                                            unsigned int stride0,
                                            unsigned int pad_en,
                                            unsigned int pad_interval_code,
                                            unsigned int pad_amount_code) {
    const unsigned long long ga = (unsigned long long)(uintptr_t)gptr;
    v4u g0;
    g0.x = 1u;                                    // count=1, user descriptor
    g0.y = lds_off;                               // lds_addr (bytes)
    g0.z = (unsigned int)ga;                      // global_addr[31:0]
    g0.w = (unsigned int)(ga >> 32) | (2u << 30); // global_addr[56:32] | type=2
    v8i g1;
    g1[0] = (int)((2u << 16) |                    // data_size = 4 bytes
                  (pad_en << 20) |
                  (pad_interval_code << 22) |
                  (pad_amount_code << 25));
    g1[1] = (int)(tensor_d0 << 16);               // tensor_dim0[15:0]
    g1[2] = (int)((tensor_d0 >> 16) | (tensor_d1 << 16));
    g1[3] = (int)((tensor_d1 >> 16) | (tile_d0 << 16));
    g1[4] = (int)tile_d1;                         // tile_dim1 (tile_dim2 = 0)
    g1[5] = (int)stride0;                         // tensor_dim0_stride[31:0]
    g1[6] = 0;
    g1[7] = 0;
    const v4i z4 = {0, 0, 0, 0};
    const v8i z8 = {0, 0, 0, 0, 0, 0, 0, 0};
    __builtin_amdgcn_tensor_load_to_lds(g0, g1, z4, z4, z8, 0);
}

// out[i] = sum_t G[(i-t) mod D, t],  G = in1^T (DxB) @ in2 (BxD), K = 128.
// Grid (64 row-blocks, 8 col-eighths), 256 threads = 8 waves, 1 tile/wave.
// LDS (dynamic): acc[1024] | aSlab[128x16] | bSlab[128x132 (TDM-padded)].
#define ACC_OFF   0
#define ASLAB_OFF 1024
#define BSLAB_OFF 3072
#define BSTRIDE   132               // 128 + 4-DWORD TDM pad -> conflict-free
#define SMEM_FLOATS (BSLAB_OFF + BATCH * BSTRIDE)

__global__ __launch_bounds__(256)
void cc_wmma_tdm_kernel(const float* __restrict__ in1,
                        const float* __restrict__ in2,
                        float* __restrict__ out) {
    extern __shared__ float smem[];
    float* acc = smem + ACC_OFF;
    float* as  = smem + ASLAB_OFF;
    float* bs  = smem + BSLAB_OFF;

    const int r0   = blockIdx.x * 16;    // row block of G
    const int cb   = blockIdx.y * 128;   // column base of this WG
    const int t    = threadIdx.x;
    const int lane = t & 31;
    const int wave = t >> 5;
    const int g    = lane >> 4;          // lane half: K pair / M offset
    const int n    = lane & 15;          // M (A frag) / N (B frag)

    if (wave == 0) {
        // A-slab: in1[:, r0:r0+16]  (tile 16 x 128 rows, stride 1024)
        tdm_load_2d((unsigned int)(uintptr_t)as, in1 + r0,
                    /*tile*/ 16, BATCH, /*tensor*/ DIM, BATCH, /*stride*/ DIM,
                    /*pad*/ 0, 0, 0);
        // B-slab: in2[:, cb:cb+128] (tile 128 x 128 rows), pad 4 DW / 128 DW
        tdm_load_2d((unsigned int)(uintptr_t)bs, in2 + cb,
                    /*tile*/ 128, BATCH, /*tensor*/ DIM, BATCH, /*stride*/ DIM,
                    /*pad*/ 1, /*interval: code 6 = 128 DW*/ 6, /*amount: code 3 = 4 DW*/ 3);
    }
#pragma unroll
    for (int i = 0; i < 4; ++i) acc[t + i * 256] = 0.0f;
    if (wave == 0) __builtin_amdgcn_s_wait_tensorcnt(0);
    __syncthreads();

    const int cl = wave * 16;            // local column block
    const int c0 = cb + cl;
    v8f c = {};
    // K = 128 in steps of 4. A frag (ISA 7.12.2): lane(g,n):
    //   v0 = A[n, k0+2g], v1 = A[n, k0+2g+1]; B mirrored with N across lanes.
#pragma unroll 4
    for (int kb = 0; kb < 32; ++kb) {
        const int k0 = kb * 4 + 2 * g;
        v2f a, b;
        a.x = as[(k0 + 0) * 16 + n];
        a.y = as[(k0 + 1) * 16 + n];
        b.x = bs[(k0 + 0) * BSTRIDE + cl + n];
        b.y = bs[(k0 + 1) * BSTRIDE + cl + n];
        c = __builtin_amdgcn_wmma_f32_16x16x4_f32(
                /*neg_a=*/false, a, /*neg_b=*/false, b,
                /*c_mod=*/(short)0, c, /*reuse_a=*/false, /*reuse_b=*/false);
    }
    // C/D layout: VGPR r, lane(g,n) holds G[r0 + r + 8g, c0 + n].
    const int dbase = r0 + c0 + 8 * g + n;
#pragma unroll
    for (int r = 0; r < 8; ++r) {
        atomicAdd(&acc[(dbase + r) & (DIM - 1)], c[r]);  // ds_add_f32
    }
    __syncthreads();

    // This WG touched d in [r0+cb, r0+cb + 16+128-2] -> 143 positions.
    const int base = r0 + cb;
    if (t < 143) {
        const int pos = (base + t) & (DIM - 1);
        atomicAdd(&out[pos], acc[pos]);  // global_atomic_add_f32
    }
}

extern "C" void kernel_launch(void* const* d_in, const int* in_sizes, int n_in,
                              void* d_out, int out_size, void* d_ws, size_t ws_size,
                              hipStream_t stream) {
    (void)in_sizes; (void)n_in; (void)out_size; (void)d_ws; (void)ws_size;
    const float* in1 = (const float*)d_in[0];
    const float* in2 = (const float*)d_in[1];
    float* out = (float*)d_out;

    cc_zero_out<<<1, 256, 0, stream>>>(out);
    cc_wmma_tdm_kernel<<<dim3(64, 8), 256, SMEM_FLOATS * sizeof(float), stream>>>(in1, in2, out);
}